// GraphAttentionLayer_33097017983377
// MI455X (gfx1250) — compile-verified
//
#include <hip/hip_runtime.h>
#include <hip/hip_bf16.h>
#include <cstdint>

typedef __attribute__((ext_vector_type(2))) float v2f;
typedef __attribute__((ext_vector_type(8))) float v8f;

#define B_    8
#define N_    1024
#define FIN_  128
#define H_    4
#define FOUT_ 64
#define HF_   256            // H_ * FOUT_
#define NEG_INF_ (-9.0e15f)

// D = A(16x4) * B(4x16) + C, full fp32 (matches reference dtype exactly).
static __device__ __forceinline__ v8f wmma_f32(v2f a, v2f b, v8f c) {
  return __builtin_amdgcn_wmma_f32_16x16x4_f32(
      /*neg_a=*/false, a, /*neg_b=*/false, b,
      /*c_mod=*/(short)0, c, /*reuse_a=*/false, /*reuse_b=*/false);
}

// ---------------- Kernel 1: h = x @ W  (8192x128 @ 128x256) ----------------
// grid(512,4) x block(128): wave w of block handles col tile (blockIdx.y*4+w).
__global__ void gat_gemm_xw(const float* __restrict__ x,
                            const float* __restrict__ W,
                            float* __restrict__ hbuf) {
  const int lane = threadIdx.x & 31;
  const int wave = threadIdx.x >> 5;
  const int row0 = blockIdx.x * 16;
  const int col0 = (blockIdx.y * 4 + wave) * 16;
  const int nl   = lane & 15;          // A row / B col within tile
  const int koff = (lane >> 4) * 2;    // half-wave K offset

  v8f acc = {};
  const float* xrow = x + (size_t)(row0 + nl) * FIN_;
  for (int k0 = 0; k0 < FIN_; k0 += 4) {
    v2f a;
    a.x = xrow[k0 + koff];
    a.y = xrow[k0 + koff + 1];
    v2f b;
    b.x = W[(size_t)(k0 + koff) * HF_ + col0 + nl];
    b.y = W[(size_t)(k0 + koff + 1) * HF_ + col0 + nl];
    acc = wmma_f32(a, b, acc);
  }
  const int half = lane >> 4;
#pragma unroll
  for (int v = 0; v < 8; ++v) {
    const int row = row0 + v + 8 * half;
    hbuf[(size_t)row * HF_ + col0 + nl] = acc[v];
  }
}

// ---------------- Kernel 2: e_src/e_dst per (b,n,h) ----------------
__global__ void gat_edges(const float* __restrict__ hbuf,
                          const float* __restrict__ a,
                          float* __restrict__ esrc,
                          float* __restrict__ edst) {
  const int tid = blockIdx.x * blockDim.x + threadIdx.x;   // 0 .. 32767
  const int hd = tid & 3;
  const int bn = tid >> 2;
  const float* hrow = hbuf + (size_t)bn * HF_ + hd * FOUT_;
  const float* asrc = a + hd * 2 * FOUT_;
  const float* adst = asrc + FOUT_;
  float s = 0.f, d = 0.f;
#pragma unroll 4
  for (int f = 0; f < FOUT_; ++f) {
    const float hv = hrow[f];
    s += hv * asrc[f];
    d += hv * adst[f];
  }
  esrc[tid] = s;
  edst[tid] = d;
}

// ---------------- Kernel 3: per-row softmax stats (max, 1/sumexp) ----------
// One wave per (b,n); all 4 heads tracked simultaneously; online max/sum,
// then wave32 shfl_xor tree reduction. Masked entries contribute NEG_INF
// (not skipped) to reproduce the reference's uniform softmax on empty rows.
__global__ void gat_rowstats(const int* __restrict__ adj,
                             const float* __restrict__ esrc,
                             const float* __restrict__ edst,
                             float* __restrict__ rmax,
                             float* __restrict__ rinv) {
  const int lane = threadIdx.x & 31;
  const int bn = blockIdx.x * (blockDim.x >> 5) + (threadIdx.x >> 5);
  const int b = bn >> 10;
  const int* adjrow = adj + (size_t)bn * N_;

  float es[H_], mx[H_], sm[H_];
#pragma unroll
  for (int hd = 0; hd < H_; ++hd) {
    es[hd] = esrc[bn * H_ + hd];
    mx[hd] = NEG_INF_;
    sm[hd] = 0.f;
  }
  for (int m = lane; m < N_; m += 32) {
    const int ad = adjrow[m];
    const float* ed = edst + ((size_t)b * N_ + m) * H_;
#pragma unroll
    for (int hd = 0; hd < H_; ++hd) {
      float e = es[hd] + ed[hd];
      e = (e > 0.f) ? e : 0.2f * e;               // leaky_relu(0.2)
      const float ev = (ad > 0) ? e : NEG_INF_;
      const float mn = fmaxf(mx[hd], ev);
      sm[hd] = sm[hd] * __expf(mx[hd] - mn) + __expf(ev - mn);
      mx[hd] = mn;
    }
  }
#pragma unroll
  for (int off = 16; off >= 1; off >>= 1) {
#pragma unroll
    for (int hd = 0; hd < H_; ++hd) {
      const float mo = __shfl_xor(mx[hd], off, 32);
      const float so = __shfl_xor(sm[hd], off, 32);
      const float mn = fmaxf(mx[hd], mo);
      sm[hd] = sm[hd] * __expf(mx[hd] - mn) + so * __expf(mo - mn);
      mx[hd] = mn;
    }
  }
  if (lane == 0) {
#pragma unroll
    for (int hd = 0; hd < H_; ++hd) {
      rmax[bn * H_ + hd] = mx[hd];
      rinv[bn * H_ + hd] = 1.0f / sm[hd];
    }
  }
}

// ---------------- Kernel 4: h' = attention @ h via f32 WMMA ----------------
// One wave per (b, head, 16-row n-tile). Per 16-wide m-tile: build attention
// A-fragments in registers, then 4 k-chunks x 4 f-tiles = 16 WMMAs.
__global__ void gat_attn_pv(const int* __restrict__ adj,
                            const float* __restrict__ hbuf,
                            const float* __restrict__ esrc,
                            const float* __restrict__ edst,
                            const float* __restrict__ rmax,
                            const float* __restrict__ rinv,
                            float* __restrict__ hp) {
  const int lane = threadIdx.x & 31;
  const int wv = blockIdx.x * (blockDim.x >> 5) + (threadIdx.x >> 5);
  const int ntile = wv & 63;
  const int hd = (wv >> 6) & 3;
  const int b = wv >> 8;
  const int n0 = ntile * 16;
  const int nl = lane & 15;
  const int koff = (lane >> 4) * 2;

  const int row = b * N_ + n0 + nl;                 // this lane's A-matrix row
  const float es_l = esrc[row * H_ + hd];
  const float mx_l = rmax[row * H_ + hd];
  const float iv_l = rinv[row * H_ + hd];
  const int* adjrow = adj + (size_t)row * N_;
  const float* vbase = hbuf + (size_t)b * N_ * HF_ + hd * FOUT_ + nl;

  v8f acc[4] = {};
  for (int m0 = 0; m0 < N_; m0 += 16) {
    v2f af[4];
#pragma unroll
    for (int kk = 0; kk < 4; ++kk) {
      const int ma = m0 + kk * 4 + koff;            // even -> aligned int2
      const int2 ad = *(const int2*)(adjrow + ma);
      const float* ed = edst + ((size_t)b * N_ + ma) * H_ + hd;
      float e0 = es_l + ed[0];
      float e1 = es_l + ed[H_];
      e0 = (e0 > 0.f) ? e0 : 0.2f * e0;
      e1 = (e1 > 0.f) ? e1 : 0.2f * e1;
      e0 = (ad.x > 0) ? e0 : NEG_INF_;
      e1 = (ad.y > 0) ? e1 : NEG_INF_;
      af[kk].x = __expf(e0 - mx_l) * iv_l;          // final attention weight
      af[kk].y = __expf(e1 - mx_l) * iv_l;
    }
#pragma unroll
    for (int ft = 0; ft < 4; ++ft) {
#pragma unroll
      for (int kk = 0; kk < 4; ++kk) {
        const int mk = m0 + kk * 4 + koff;
        const float* vp = vbase + (size_t)mk * HF_ + ft * 16;
        v2f bf;
        bf.x = vp[0];
        bf.y = vp[HF_];
        acc[ft] = wmma_f32(af[kk], bf, acc[ft]);
      }
    }
  }
  const int half = lane >> 4;
#pragma unroll
  for (int ft = 0; ft < 4; ++ft) {
#pragma unroll
    for (int v = 0; v < 8; ++v) {
      const int n = n0 + v + 8 * half;
      hp[(((size_t)b * N_ + n) * H_ + hd) * FOUT_ + ft * 16 + nl] = acc[ft][v];
    }
  }
}

// ---------------- Kernel 5: mean over heads -> d_out ----------------
__global__ void gat_mean(const float* __restrict__ hp, float* __restrict__ out) {
  const int tid = blockIdx.x * blockDim.x + threadIdx.x;  // 0 .. 524287
  const int f = tid & 63;
  const int bn = tid >> 6;
  const float* p = hp + (size_t)bn * H_ * FOUT_ + f;
  out[tid] = 0.25f * (p[0] + p[FOUT_] + p[2 * FOUT_] + p[3 * FOUT_]);
}

extern "C" void kernel_launch(void* const* d_in, const int* in_sizes, int n_in,
                              void* d_out, int out_size, void* d_ws, size_t ws_size,
                              hipStream_t stream) {
  (void)in_sizes; (void)n_in; (void)out_size; (void)ws_size;
  const float* x   = (const float*)d_in[0];
  const int*   adj = (const int*)d_in[1];
  const float* W   = (const float*)d_in[2];
  const float* a   = (const float*)d_in[3];
  float* out = (float*)d_out;
  float* ws  = (float*)d_ws;

  float* hbuf = ws;                                   // 2,097,152 floats (8 MB)
  float* esrc = hbuf + (size_t)B_ * N_ * HF_;         //    32,768
  float* edst = esrc + B_ * N_ * H_;                  //    32,768
  float* rmax = edst + B_ * N_ * H_;                  //    32,768
  float* rinv = rmax + B_ * N_ * H_;                  //    32,768
  float* hp   = rinv + B_ * N_ * H_;                  // 2,097,152 (8 MB)

  gat_gemm_xw <<<dim3(512, 4), 128, 0, stream>>>(x, W, hbuf);
  gat_edges   <<<128,  256, 0, stream>>>(hbuf, a, esrc, edst);
  gat_rowstats<<<2048, 128, 0, stream>>>(adj, esrc, edst, rmax, rinv);
  gat_attn_pv <<<512,  128, 0, stream>>>(adj, hbuf, esrc, edst, rmax, rinv, hp);
  gat_mean    <<<2048, 256, 0, stream>>>(hp, out);
}